// ProbabilityAdjustedLoss_52819507806483
// MI455X (gfx1250) — compile-verified
//
#include <hip/hip_runtime.h>
#include <hip/hip_bf16.h>

// ProbabilityAdjustedLoss — B=8388608 rows x 2 logits, int labels, scalar mean loss.
// Memory-bound streaming reduction (~100MB @ 23.3TB/s ≈ 4.3us floor).
// Wave-level reduction routed through V_WMMA_F32_16X16X4_F32 (B = ones),
// cross-half combine via ds_swizzle SWAPX16, inter-wave combine serial in lane 0.

typedef float v2f __attribute__((ext_vector_type(2)));
typedef float v4f __attribute__((ext_vector_type(4)));
typedef float v8f __attribute__((ext_vector_type(8)));
typedef int   v4i __attribute__((ext_vector_type(4)));

// Full-wave (32-lane) sum, result broadcast to all lanes. Requires EXEC all-1s.
// WMMA D[m][n] = sum_k A[m][k]*1: with A vgpr0 = laneSum, vgpr1 = 0 the ISA
// A-layout gives D[m][n] = laneSum[m] + laneSum[m+16]. Each lane holds 8 rows
// of D (lanes 0-15: M=0..7, lanes 16-31: M=8..15), so per-lane sum of the 8
// accumulators + one SWAPX16 swizzle-add = full 32-lane sum in every lane.
__device__ __forceinline__ float wave_allsum(float v) {
#if defined(__gfx1250__) && __has_builtin(__builtin_amdgcn_wmma_f32_16x16x4_f32)
    v2f a; a[0] = v;    a[1] = 0.0f;
    v2f b; b[0] = 1.0f; b[1] = 1.0f;
    v8f c = {};
    c = __builtin_amdgcn_wmma_f32_16x16x4_f32(
        /*neg_a=*/false, a, /*neg_b=*/false, b,
        /*c_mod=*/(short)0, c, /*reuse_a=*/false, /*reuse_b=*/false);
    float s = ((c[0] + c[1]) + (c[2] + c[3])) + ((c[4] + c[5]) + (c[6] + c[7]));
    // SWAPX16: group-of-32 swizzle, xor_mask=0x10, and_mask=0x1f -> 0x401f
    s += __int_as_float(__builtin_amdgcn_ds_swizzle(__float_as_int(s), 0x401f));
    return s;
#else
    for (int off = 16; off > 0; off >>= 1) v += __shfl_xor(v, off, 32);
    return v;
#endif
}

// Per-row loss. p0 = 1/(1+e^(l1-l0)) == softmax(l0); branchless select by label.
__device__ __forceinline__ float row_loss(float l0, float l1, int lab) {
    float e  = __expf(l1 - l0);
    float p0 = __builtin_amdgcn_rcpf(1.0f + e);
    float p1 = 1.0f - p0;
    // w0 = max(0.1, 1 - max(0, p0-0.95)/(1-0.95)*0.5) = max(0.1, 1 - max(0,p0-0.95)*10)
    float w0 = fmaxf(0.1f, 1.0f - fmaxf(0.0f, p0 - 0.95f) * 10.0f);
    // w1 = max(0.5, 2 - max(0, p1-0.05)/0.95)
    float w1 = fmaxf(0.5f, 2.0f - fmaxf(0.0f, p1 - 0.05f) * (1.0f / 0.95f));
    bool stable = (lab == 0);
    float p = stable ? p0 : p1;
    float w = stable ? w0 : w1;
    return -w * __logf(p + 1e-8f);
}

// Main pass: each iteration consumes 4 rows (two B128 logit loads + one B128
// label load), grid-stride. Per-block partial written to d_ws. Blocks are
// fixed at 256 threads = 8 wave32.
__global__ void pal_main(const v4f* __restrict__ lg4,
                         const v4i* __restrict__ lb4,
                         float* __restrict__ partials,
                         int ngroups) {
    int tid    = blockIdx.x * blockDim.x + threadIdx.x;
    int stride = gridDim.x * blockDim.x;
    float acc = 0.0f;
    for (int i = tid; i < ngroups; i += stride) {
        v4f ab = __builtin_nontemporal_load(&lg4[2 * i]);      // rows 4i, 4i+1
        v4f cd = __builtin_nontemporal_load(&lg4[2 * i + 1]);  // rows 4i+2, 4i+3
        v4i lb = __builtin_nontemporal_load(&lb4[i]);
        acc += row_loss(ab[0], ab[1], lb[0]);
        acc += row_loss(ab[2], ab[3], lb[1]);
        acc += row_loss(cd[0], cd[1], lb[2]);
        acc += row_loss(cd[2], cd[3], lb[3]);
    }
    float s = wave_allsum(acc);

    __shared__ float smem[8];
    int lane = threadIdx.x & 31;
    int wv   = threadIdx.x >> 5;
    if (lane == 0) smem[wv] = s;
    __syncthreads();
    if (threadIdx.x == 0) {
        float v = smem[0];
#pragma unroll
        for (int j = 1; j < 8; ++j) v += smem[j];
        partials[blockIdx.x] = v;
    }
}

// Finalize: one 256-thread block deterministically reduces the block partials
// (uniform trip count -> EXEC stays all-1s for the WMMA reduce), scales by 1/B.
__global__ void pal_finalize(const float* __restrict__ partials, int n,
                             float* __restrict__ out, float scale) {
    float acc = 0.0f;
    for (int i = threadIdx.x; i < n; i += blockDim.x) acc += partials[i];
    float s = wave_allsum(acc);

    __shared__ float smem[8];
    int lane = threadIdx.x & 31;
    int wv   = threadIdx.x >> 5;
    if (lane == 0) smem[wv] = s;
    __syncthreads();
    if (threadIdx.x == 0) {
        float v = smem[0];
#pragma unroll
        for (int j = 1; j < 8; ++j) v += smem[j];
        out[0] = v * scale;  // TASK_WEIGHT == 1.0
    }
}

extern "C" void kernel_launch(void* const* d_in, const int* in_sizes, int n_in,
                              void* d_out, int out_size, void* d_ws, size_t ws_size,
                              hipStream_t stream) {
    const float* logits = (const float*)d_in[0];  // (B, 2) fp32
    const int*   labels = (const int*)d_in[1];    // (B,) int
    float*       out    = (float*)d_out;          // scalar fp32
    float*       parts  = (float*)d_ws;           // >= GRID floats

    long rows    = (long)in_sizes[1];             // B = 8388608
    int  ngroups = (int)(rows / 4);               // 4 rows per group

    const int BLK  = 256;                         // 8 wave32 per block
    const int GRID = 2048;                        // 16384 waves in flight

    pal_main<<<GRID, BLK, 0, stream>>>((const v4f*)logits, (const v4i*)labels,
                                       parts, ngroups);
    pal_finalize<<<1, BLK, 0, stream>>>(parts, GRID, out, 1.0f / (float)rows);
}